// SimpleLSTM_4011499454656
// MI455X (gfx1250) — compile-verified
//
#include <hip/hip_runtime.h>
#include <hip/hip_bf16.h>
#include <math.h>

// Problem sizes (fixed by the reference).
#define N_SEQ 8192
#define DIN   1024
#define HD    1024
#define H4    4096

typedef float v2f __attribute__((ext_vector_type(2)));
typedef float v8f __attribute__((ext_vector_type(8)));

// ---------------- workspace layout ----------------
// [0, 128MB)       : x_gates fp32 [N_SEQ][H4]
// [+0, +8KB)       : h double buffer (2 * HD floats)
// [+8KB, +8KB+8B)  : barrier {arrive, generation}
#define XG_BYTES ((size_t)N_SEQ * (size_t)H4 * sizeof(float))

// LDS byte offset (wave-relative) of a __shared__ object, for async-to-LDS VDST.
__device__ __forceinline__ unsigned lds_off(const void* p) {
  return (unsigned)(__UINTPTR_TYPE__)(__attribute__((address_space(3))) const void*)p;
}

// ---------------- init: seed h, reset barrier ----------------
__global__ void lstm_init_kernel(const float* __restrict__ h0,
                                 float* __restrict__ hbuf,
                                 unsigned* __restrict__ bar) {
  int i = blockIdx.x * blockDim.x + threadIdx.x;
  if (i < HD) hbuf[i] = h0[i];          // buffer 0 = h_{t=0}
  if (i == 0) bar[0] = 0u;              // arrive counter; bar[1] (gen) may be anything
}

// ---------------- phase 1: x_gates = xs @ W_ih^T + (b_ih + b_hh) ----------------
// fp32 WMMA 16x16x4. Each wave computes a 32(M) x 64(N) strip: 2x4 register
// tile -> 8 wmma per 6 fragment loads per k-step. Block = 8 waves (2M x 4N)
// -> 64(M) x 256(N) block tile.
__global__ __launch_bounds__(256)
void lstm_xgemm_wmma(const float* __restrict__ xs,
                     const float* __restrict__ Wih,
                     const float* __restrict__ bih,
                     const float* __restrict__ bhh,
                     float* __restrict__ xg) {
  const int lane    = threadIdx.x & 31;
  const int wave    = threadIdx.x >> 5;
  const int lane_lo = lane & 15;        // M (for A) / N (for B/C/D)
  const int lane_hi = lane >> 4;        // selects K half (A/B), M half (C/D)

  const int row0 = blockIdx.y * 64 + (wave >> 2) * 32;   // 2 M-tiles
  const int col0 = blockIdx.x * 256 + (wave & 3) * 64;   // 4 N-tiles

  const float* a0r = xs  + (size_t)(row0 +  0 + lane_lo) * DIN;
  const float* a1r = xs  + (size_t)(row0 + 16 + lane_lo) * DIN;
  const float* b0r = Wih + (size_t)(col0 +  0 + lane_lo) * DIN;
  const float* b1r = Wih + (size_t)(col0 + 16 + lane_lo) * DIN;
  const float* b2r = Wih + (size_t)(col0 + 32 + lane_lo) * DIN;
  const float* b3r = Wih + (size_t)(col0 + 48 + lane_lo) * DIN;

  v8f acc00 = {}, acc01 = {}, acc02 = {}, acc03 = {};
  v8f acc10 = {}, acc11 = {}, acc12 = {}, acc13 = {};

#pragma unroll 2
  for (int k = 0; k < DIN; k += 4) {
    const int ko = k + 2 * lane_hi;     // ISA A 16x4 layout: lanes16-31 hold K=2,3
    v2f a0 = *(const v2f*)(a0r + ko);
    v2f a1 = *(const v2f*)(a1r + ko);
    v2f b0 = *(const v2f*)(b0r + ko);
    v2f b1 = *(const v2f*)(b1r + ko);
    v2f b2 = *(const v2f*)(b2r + ko);
    v2f b3 = *(const v2f*)(b3r + ko);
    acc00 = __builtin_amdgcn_wmma_f32_16x16x4_f32(false, a0, false, b0, (short)0, acc00, false, false);
    acc01 = __builtin_amdgcn_wmma_f32_16x16x4_f32(false, a0, false, b1, (short)0, acc01, false, false);
    acc02 = __builtin_amdgcn_wmma_f32_16x16x4_f32(false, a0, false, b2, (short)0, acc02, false, false);
    acc03 = __builtin_amdgcn_wmma_f32_16x16x4_f32(false, a0, false, b3, (short)0, acc03, false, false);
    acc10 = __builtin_amdgcn_wmma_f32_16x16x4_f32(false, a1, false, b0, (short)0, acc10, false, false);
    acc11 = __builtin_amdgcn_wmma_f32_16x16x4_f32(false, a1, false, b1, (short)0, acc11, false, false);
    acc12 = __builtin_amdgcn_wmma_f32_16x16x4_f32(false, a1, false, b2, (short)0, acc12, false, false);
    acc13 = __builtin_amdgcn_wmma_f32_16x16x4_f32(false, a1, false, b3, (short)0, acc13, false, false);
  }

  // C/D layout: VGPR r -> M = row0 + 16*m + 8*lane_hi + r, N = col0 + 16*sub + lane_lo
#pragma unroll
  for (int m = 0; m < 2; ++m) {
    const int mbase = row0 + 16 * m + 8 * lane_hi;
#pragma unroll
    for (int sub = 0; sub < 4; ++sub) {
      const int col = col0 + 16 * sub + lane_lo;
      const float bias = bih[col] + bhh[col];
      v8f acc;
      if (m == 0) acc = (sub == 0) ? acc00 : (sub == 1) ? acc01 : (sub == 2) ? acc02 : acc03;
      else        acc = (sub == 0) ? acc10 : (sub == 1) ? acc11 : (sub == 2) ? acc12 : acc13;
#pragma unroll
      for (int r = 0; r < 8; ++r) {
        xg[(size_t)(mbase + r) * H4 + col] = acc[r] + bias;
      }
    }
  }
}

// ---------------- device-wide barrier for the persistent scan ----------------
__device__ __forceinline__ void grid_sync(unsigned* bar, int nblk) {
  __syncthreads();
  if (threadIdx.x == 0) {
    __threadfence();                                  // publish this block's h writes
    unsigned gen  = __hip_atomic_load(bar + 1, __ATOMIC_ACQUIRE, __HIP_MEMORY_SCOPE_AGENT);
    unsigned prev = __hip_atomic_fetch_add(bar, 1u, __ATOMIC_ACQ_REL, __HIP_MEMORY_SCOPE_AGENT);
    if (prev == (unsigned)(nblk - 1)) {
      __hip_atomic_store(bar, 0u, __ATOMIC_RELAXED, __HIP_MEMORY_SCOPE_AGENT);
      __hip_atomic_fetch_add(bar + 1, 1u, __ATOMIC_RELEASE, __HIP_MEMORY_SCOPE_AGENT);
    } else {
      while (__hip_atomic_load(bar + 1, __ATOMIC_ACQUIRE, __HIP_MEMORY_SCOPE_AGENT) == gen)
        __builtin_amdgcn_s_sleep(1);
    }
  }
  __syncthreads();
}

// ---------------- phase 2: sequential LSTM scan ----------------
// 64 blocks x 256 threads. Block b owns h indices j0..j0+15 (j0 = 16*b),
// i.e. W_hh rows {g*HD + j0 + jj : g in 0..3, jj in 0..15} -> 64 rows.
// 4 threads per row, each a 256-length partial dot over h staged in LDS via
// GLOBAL_LOAD_ASYNC_TO_LDS_B128 (ASYNCcnt). c lives in registers of threads
// 0..15; h is double-buffered in d_ws (one device barrier per step).
#define RBLK 64
__global__ __launch_bounds__(256)
void lstm_recur(const float* __restrict__ Whh,
                const float* __restrict__ xg,
                const float* __restrict__ c0,
                float* __restrict__ hbuf,      // 2 * HD floats
                float* __restrict__ ys,
                unsigned* __restrict__ bar) {
  __shared__ __align__(16) float h_lds[HD];
  __shared__ float g_lds[64];

  const int tid        = threadIdx.x;
  const int row_in_blk = tid >> 2;          // 0..63
  const int part       = tid & 3;           // 0..3 (quad within a wave)
  const int j0         = blockIdx.x * 16;
  const int gate       = row_in_blk >> 4;   // 0:i 1:f 2:g 3:o (PyTorch order)
  const int jj         = row_in_blk & 15;
  const int grow       = gate * HD + j0 + jj;

  const float* wrow = Whh + (size_t)grow * HD + part * 256;
  const unsigned h_dst = lds_off(&h_lds[0]) + (unsigned)tid * 16u;  // 16B per thread

  float creg = 0.f;
  if (tid < 16) creg = c0[j0 + tid];

  for (int t = 0; t < N_SEQ; ++t) {
    const float* hcur = hbuf + (size_t)(t & 1) * HD;
    float*       hnxt = hbuf + (size_t)((t + 1) & 1) * HD;

    // stage h_t into LDS: async copy, 16 bytes/thread, device scope (h was
    // produced by other WGPs last step).
    {
      unsigned long long gsrc = (unsigned long long)(const void*)hcur
                              + (unsigned long long)tid * 16ull;
      asm volatile("global_load_async_to_lds_b128 %0, %1, off scope:SCOPE_DEV"
                   :: "v"(h_dst), "v"(gsrc) : "memory");
      asm volatile("s_wait_asynccnt 0x0" ::: "memory");
    }
    __syncthreads();

    // partial dot over 256 elements (float4 vectorized)
    float acc = 0.f;
    const float* hp = h_lds + part * 256;
#pragma unroll 8
    for (int k = 0; k < 256; k += 4) {
      float4 w = *(const float4*)(wrow + k);
      float4 hv = *(const float4*)(hp + k);
      acc = fmaf(w.x, hv.x, acc);
      acc = fmaf(w.y, hv.y, acc);
      acc = fmaf(w.z, hv.z, acc);
      acc = fmaf(w.w, hv.w, acc);
    }
    // reduce quad -> full dot; add precomputed x-gate
    acc += __shfl_xor(acc, 1, 32);
    acc += __shfl_xor(acc, 2, 32);
    if (part == 0) g_lds[row_in_blk] = acc + xg[(size_t)t * H4 + grow];
    __syncthreads();

    // cell update, local to this block's 16 h indices
    if (tid < 16) {
      const int j = j0 + tid;
      const float gi = g_lds[ 0 + tid];
      const float gf = g_lds[16 + tid];
      const float gg = g_lds[32 + tid];
      const float go = g_lds[48 + tid];
      const float ig = 1.f / (1.f + __expf(-gi));
      const float fg = 1.f / (1.f + __expf(-gf));
      const float gv = tanhf(gg);
      const float og = 1.f / (1.f + __expf(-go));
      creg = fg * creg + ig * gv;
      const float hn = og * tanhf(creg);
      hnxt[j] = hn;
      ys[(size_t)t * HD + j] = hn;
    }

    grid_sync(bar, RBLK);                   // one device barrier per step
  }
}

extern "C" void kernel_launch(void* const* d_in, const int* in_sizes, int n_in,
                              void* d_out, int out_size, void* d_ws, size_t ws_size,
                              hipStream_t stream) {
  (void)in_sizes; (void)n_in; (void)out_size; (void)ws_size;
  const float* xs  = (const float*)d_in[0];
  const float* Wih = (const float*)d_in[1];
  const float* Whh = (const float*)d_in[2];
  const float* bih = (const float*)d_in[3];
  const float* bhh = (const float*)d_in[4];
  const float* h0  = (const float*)d_in[5];
  const float* c0  = (const float*)d_in[6];
  float* ys = (float*)d_out;

  float*    xg   = (float*)d_ws;
  float*    hbuf = (float*)((char*)d_ws + XG_BYTES);
  unsigned* bar  = (unsigned*)((char*)d_ws + XG_BYTES + 2 * HD * sizeof(float));

  lstm_init_kernel<<<4, 256, 0, stream>>>(h0, hbuf, bar);

  dim3 g1(H4 / 256, N_SEQ / 64);            // (16, 128) blocks of 256 threads
  lstm_xgemm_wmma<<<g1, 256, 0, stream>>>(xs, Wih, bih, bhh, xg);

  lstm_recur<<<RBLK, 256, 0, stream>>>(Whh, xg, c0, hbuf, ys, bar);
}